// Model_44538810860092
// MI455X (gfx1250) — compile-verified
//
#include <hip/hip_runtime.h>

typedef float v2f __attribute__((ext_vector_type(2)));
typedef float v8f __attribute__((ext_vector_type(8)));

// ---------------------------------------------------------------------------
// WMMA fp32 strip-GEMM: C[M,N] = A[M,K] * B[K,N]
//   - one wave computes a full 16-row x N strip (TILES_N = N/16 accumulators)
//   - A loaded once per k-step (b64), shared across all N-tiles in registers
//   - V_WMMA_F32_16X16X4_F32 (CDNA5 wave32); M%16==0, K%4==0, N==TILES_N*16
// VGPR layouts per ISA 7.12.2:
//   A 16x4 : lanes 0-15 -> M=lane, VGPR{0,1}=K{0,1}; lanes 16-31 -> K{2,3}
//   B 4x16 : lanes 0-15 -> N=lane, VGPR{0,1}=K{0,1}; lanes 16-31 -> K{2,3}
//   C 16x16: VGPR r -> row r (lanes 0-15) / row r+8 (lanes 16-31), col=lane&15
// ---------------------------------------------------------------------------
template <int TILES_N>
__global__ __launch_bounds__(128) void gemm_wmma_f32_strip(
    const float* __restrict__ A, const float* __restrict__ B,
    float* __restrict__ C, int M, int K) {
  constexpr int N = TILES_N * 16;
  const int wid  = threadIdx.x >> 5;                 // wave in block (0..3)
  const int tm   = blockIdx.x * 4 + wid;             // 16-row strip index
  const int Mt   = M >> 4;
  if (tm >= Mt) return;                              // uniform per-wave exit

  const int lane = threadIdx.x & 31;
  const int hi   = lane >> 4;                        // half-wave selector
  const int lo   = lane & 15;

  const float* Arow = A + (size_t)(tm * 16 + lo) * K + 2 * hi;
  const float* Bp0  = B + (size_t)(2 * hi) * N + lo;

  v8f acc[TILES_N];
#pragma unroll
  for (int t = 0; t < TILES_N; ++t) acc[t] = (v8f){};

  for (int k = 0; k < K; k += 4) {
    v2f a;
    a.x = Arow[k];
    a.y = Arow[k + 1];
    __builtin_prefetch(Arow + k + 32, 0, 0);         // stream next A line
    const float* Bp = Bp0 + (size_t)k * N;
#pragma unroll
    for (int t = 0; t < TILES_N; ++t) {
      v2f b;
      b.x = Bp[t * 16];
      b.y = Bp[t * 16 + N];
      acc[t] = __builtin_amdgcn_wmma_f32_16x16x4_f32(
          /*neg_a=*/false, a, /*neg_b=*/false, b,
          /*c_mod=*/(short)0, acc[t], /*reuse_a=*/false, /*reuse_b=*/false);
    }
  }

  float* Crow = C + (size_t)(tm * 16 + 8 * hi) * N + lo;
#pragma unroll
  for (int t = 0; t < TILES_N; ++t)
#pragma unroll
    for (int r = 0; r < 8; ++r) Crow[(size_t)r * N + t * 16] = acc[t][r];
}

// ---------------------------------------------------------------------------
// Utility kernels
// ---------------------------------------------------------------------------
__global__ void zero_f32(float* __restrict__ p, long long n) {
  long long i = (long long)blockIdx.x * blockDim.x + threadIdx.x;
  long long stride = (long long)gridDim.x * blockDim.x;
  for (; i < n; i += stride) p[i] = 0.0f;
}

__global__ void degree_kernel(const int* __restrict__ dst, float* __restrict__ deg, int E) {
  int i = blockIdx.x * blockDim.x + threadIdx.x;
  int stride = gridDim.x * blockDim.x;
  for (; i < E; i += stride) atomicAdd(&deg[dst[i]], 1.0f);
}

__global__ void recip_clamp_kernel(float* __restrict__ deg, int n) {
  int i = blockIdx.x * blockDim.x + threadIdx.x;
  int stride = gridDim.x * blockDim.x;
  for (; i < n; i += stride) deg[i] = 1.0f / fmaxf(deg[i], 1.0f);
}

// agg[dst[e], f] += msg[src[e], f] * rdeg[dst[e]]   (mean fused into scatter)
__global__ void scatter_mean_kernel(const float* __restrict__ msg,
                                    const int* __restrict__ src,
                                    const int* __restrict__ dst,
                                    const float* __restrict__ rdeg,
                                    float* __restrict__ agg,
                                    int E, int logF) {
  const long long total = ((long long)E) << logF;
  const int Fm1 = (1 << logF) - 1;
  long long i = (long long)blockIdx.x * blockDim.x + threadIdx.x;
  long long stride = (long long)gridDim.x * blockDim.x;
  for (; i < total; i += stride) {
    const int e = (int)(i >> logF);
    const int f = (int)(i & Fm1);
    const int s = src[e];
    const int d = dst[e];
    atomicAdd(&agg[(((long long)d) << logF) + f],
              msg[(((long long)s) << logF) + f] * rdeg[d]);
  }
}

// out = (relu?)(agg + self + bias[f]); n = rows*F
__global__ void combine_kernel(const float* __restrict__ agg,
                               const float* __restrict__ self,
                               const float* __restrict__ bias,
                               float* __restrict__ out,
                               long long n, int logF, int do_relu) {
  const int Fm1 = (1 << logF) - 1;
  long long i = (long long)blockIdx.x * blockDim.x + threadIdx.x;
  long long stride = (long long)gridDim.x * blockDim.x;
  for (; i < n; i += stride) {
    float v = agg[i] + self[i] + bias[(int)(i & Fm1)];
    out[i] = do_relu ? fmaxf(v, 0.0f) : v;
  }
}

// Diagonal-bilinear scorer: out[e] = sum_f h[src[e],f] * wrel[f] * h[dst[e],f]
// EMB == 32 == wave32: one wave per edge, lane = feature.
__global__ void score_kernel(const float* __restrict__ h,
                             const int* __restrict__ src,
                             const int* __restrict__ dst,
                             const float* __restrict__ wrel,
                             float* __restrict__ out, int E) {
  const int wave = (int)((blockIdx.x * (long long)blockDim.x + threadIdx.x) >> 5);
  const int lane = threadIdx.x & 31;
  if (wave >= E) return;
  const int s = src[wave];
  const int d = dst[wave];
  float v = h[((long long)s << 5) + lane] * wrel[lane] * h[((long long)d << 5) + lane];
#pragma unroll
  for (int off = 16; off > 0; off >>= 1) v += __shfl_down(v, off, 32);
  if (lane == 0) out[wave] = v;
}

// ---------------------------------------------------------------------------
// Orchestration
// ---------------------------------------------------------------------------
static inline void launch_gemm(const float* A, const float* B, float* C,
                               int M, int K, int N, hipStream_t stream) {
  const int Mt = M / 16;
  dim3 grid((unsigned)((Mt + 3) / 4));
  if (N == 64) {
    gemm_wmma_f32_strip<4><<<grid, 128, 0, stream>>>(A, B, C, M, K);
  } else {  // N == 32
    gemm_wmma_f32_strip<2><<<grid, 128, 0, stream>>>(A, B, C, M, K);
  }
}

extern "C" void kernel_launch(void* const* d_in, const int* in_sizes, int n_in,
                              void* d_out, int out_size, void* d_ws, size_t ws_size,
                              hipStream_t stream) {
  const float* feat_drug = (const float*)d_in[0];
  const float* feat_gene = (const float*)d_in[1];
  const int* dd_src = (const int*)d_in[2];
  const int* dd_dst = (const int*)d_in[3];
  const int* dg_src = (const int*)d_in[4];
  const int* dg_dst = (const int*)d_in[5];
  const int* gg_src = (const int*)d_in[6];
  const int* gg_dst = (const int*)d_in[7];
  const int* neg_src = (const int*)d_in[8];
  const int* neg_dst = (const int*)d_in[9];
  const float* W1_dd = (const float*)d_in[10];
  const float* W1_dg = (const float*)d_in[11];
  const float* W1_gg = (const float*)d_in[12];
  const float* W1_sd = (const float*)d_in[13];
  const float* W1_sg = (const float*)d_in[14];
  const float* b1    = (const float*)d_in[15];
  const float* W2_dd = (const float*)d_in[16];
  const float* W2_dg = (const float*)d_in[17];
  const float* W2_gg = (const float*)d_in[18];
  const float* W2_sd = (const float*)d_in[19];
  const float* W2_sg = (const float*)d_in[20];
  const float* b2    = (const float*)d_in[21];
  const float* w_rel = (const float*)d_in[22];

  const int D_DRUG = 128, D_GENE = 64, HID = 64, EMB = 32;
  const int ND = in_sizes[0] / D_DRUG;   // 20000
  const int NG = in_sizes[1] / D_GENE;   // 40000
  const int E_DD = in_sizes[2];
  const int E_DG = in_sizes[4];
  const int E_GG = in_sizes[6];
  const int E_NEG = in_sizes[8];

  // ---- workspace layout (floats); L2 buffers reuse L1 regions (half width) --
  float* ws = (float*)d_ws;
  size_t o = 0;
  float* rdeg_dd = ws + o; o += (size_t)ND;
  float* rdeg_dg = ws + o; o += (size_t)NG;
  float* rdeg_gg = ws + o; o += (size_t)NG;
  float* msg_dd  = ws + o; o += (size_t)ND * HID;
  float* msg_dg  = ws + o; o += (size_t)ND * HID;
  float* msg_gg  = ws + o; o += (size_t)NG * HID;
  float* self_d  = ws + o; o += (size_t)ND * HID;
  float* self_g  = ws + o; o += (size_t)NG * HID;
  float* agg_d   = ws + o; o += (size_t)ND * HID;
  float* agg_g   = ws + o; o += (size_t)NG * HID;
  float* h1_d    = ws + o; o += (size_t)ND * HID;
  float* h1_g    = ws + o; o += (size_t)NG * HID;
  float* h2_d    = ws + o; o += (size_t)ND * EMB;
  float* h2_g    = ws + o; o += (size_t)NG * EMB;
  (void)ws_size; (void)n_in; (void)out_size;

  const int TB = 256;
  auto blocks = [](long long n, int tb) -> unsigned {
    long long b = (n + tb - 1) / tb;
    return (unsigned)(b > 0 ? b : 1);
  };

  // ---- degrees (shared by both layers) ----
  long long degN = (long long)ND + 2LL * NG;
  zero_f32<<<blocks(degN, TB), TB, 0, stream>>>(rdeg_dd, degN);
  degree_kernel<<<blocks(E_DD, TB), TB, 0, stream>>>(dd_dst, rdeg_dd, E_DD);
  degree_kernel<<<blocks(E_DG, TB), TB, 0, stream>>>(dg_dst, rdeg_dg, E_DG);
  degree_kernel<<<blocks(E_GG, TB), TB, 0, stream>>>(gg_dst, rdeg_gg, E_GG);
  recip_clamp_kernel<<<blocks(ND, TB), TB, 0, stream>>>(rdeg_dd, ND);
  recip_clamp_kernel<<<blocks(NG, TB), TB, 0, stream>>>(rdeg_dg, NG);
  recip_clamp_kernel<<<blocks(NG, TB), TB, 0, stream>>>(rdeg_gg, NG);

  // ======================= Layer 1 (D -> HID, relu) =======================
  launch_gemm(feat_drug, W1_dd, msg_dd, ND, D_DRUG, HID, stream);
  launch_gemm(feat_drug, W1_dg, msg_dg, ND, D_DRUG, HID, stream);
  launch_gemm(feat_gene, W1_gg, msg_gg, NG, D_GENE, HID, stream);
  launch_gemm(feat_drug, W1_sd, self_d, ND, D_DRUG, HID, stream);
  launch_gemm(feat_gene, W1_sg, self_g, NG, D_GENE, HID, stream);

  zero_f32<<<blocks((long long)ND * HID, TB), TB, 0, stream>>>(agg_d, (long long)ND * HID);
  zero_f32<<<blocks((long long)NG * HID, TB), TB, 0, stream>>>(agg_g, (long long)NG * HID);

  scatter_mean_kernel<<<blocks((long long)E_DD * HID, TB), TB, 0, stream>>>(
      msg_dd, dd_src, dd_dst, rdeg_dd, agg_d, E_DD, 6);
  scatter_mean_kernel<<<blocks((long long)E_DG * HID, TB), TB, 0, stream>>>(
      msg_dg, dg_src, dg_dst, rdeg_dg, agg_g, E_DG, 6);
  scatter_mean_kernel<<<blocks((long long)E_GG * HID, TB), TB, 0, stream>>>(
      msg_gg, gg_src, gg_dst, rdeg_gg, agg_g, E_GG, 6);

  combine_kernel<<<blocks((long long)ND * HID, TB), TB, 0, stream>>>(
      agg_d, self_d, b1, h1_d, (long long)ND * HID, 6, 1);
  combine_kernel<<<blocks((long long)NG * HID, TB), TB, 0, stream>>>(
      agg_g, self_g, b1, h1_g, (long long)NG * HID, 6, 1);

  // ======================= Layer 2 (HID -> EMB) ==========================
  // reuse msg/self/agg buffers (EMB < HID so regions are large enough)
  launch_gemm(h1_d, W2_dd, msg_dd, ND, HID, EMB, stream);
  launch_gemm(h1_d, W2_dg, msg_dg, ND, HID, EMB, stream);
  launch_gemm(h1_g, W2_gg, msg_gg, NG, HID, EMB, stream);
  launch_gemm(h1_d, W2_sd, self_d, ND, HID, EMB, stream);
  launch_gemm(h1_g, W2_sg, self_g, NG, HID, EMB, stream);

  zero_f32<<<blocks((long long)ND * EMB, TB), TB, 0, stream>>>(agg_d, (long long)ND * EMB);
  zero_f32<<<blocks((long long)NG * EMB, TB), TB, 0, stream>>>(agg_g, (long long)NG * EMB);

  scatter_mean_kernel<<<blocks((long long)E_DD * EMB, TB), TB, 0, stream>>>(
      msg_dd, dd_src, dd_dst, rdeg_dd, agg_d, E_DD, 5);
  scatter_mean_kernel<<<blocks((long long)E_DG * EMB, TB), TB, 0, stream>>>(
      msg_dg, dg_src, dg_dst, rdeg_dg, agg_g, E_DG, 5);
  scatter_mean_kernel<<<blocks((long long)E_GG * EMB, TB), TB, 0, stream>>>(
      msg_gg, gg_src, gg_dst, rdeg_gg, agg_g, E_GG, 5);

  combine_kernel<<<blocks((long long)ND * EMB, TB), TB, 0, stream>>>(
      agg_d, self_d, b2, h2_d, (long long)ND * EMB, 5, 0);
  combine_kernel<<<blocks((long long)NG * EMB, TB), TB, 0, stream>>>(
      agg_g, self_g, b2, h2_g, (long long)NG * EMB, 5, 0);
  (void)h2_g;  // gene embeddings not scored, but computed per reference

  // ======================= Scorer (wave per edge) ========================
  float* out_pos = (float*)d_out;
  float* out_neg = out_pos + E_DD;
  score_kernel<<<blocks((long long)E_DD * 32, TB), TB, 0, stream>>>(
      h2_d, dd_src, dd_dst, w_rel, out_pos, E_DD);
  score_kernel<<<blocks((long long)E_NEG * 32, TB), TB, 0, stream>>>(
      h2_d, neg_src, neg_dst, w_rel, out_neg, E_NEG);
}